// JKNet_3layer_17205638988376
// MI455X (gfx1250) — compile-verified
//
#include <hip/hip_runtime.h>

#define NN    50000
#define NE    800000
#define DIM   128
#define NCLS  32
#define NGR   64
#define BN_EPS 1e-5f

typedef float v2f __attribute__((ext_vector_type(2)));
typedef float v8f __attribute__((ext_vector_type(8)));

// ---------------- utility kernels ----------------

__global__ void zero_f32(float* __restrict__ p, long n) {
    long i = (long)blockIdx.x * blockDim.x + threadIdx.x;
    if (i < n) p[i] = 0.f;
}

__global__ void degree_kernel(const int* __restrict__ dst, float* __restrict__ deg, int E) {
    int e = blockIdx.x * blockDim.x + threadIdx.x;
    if (e < E) atomicAdd(&deg[dst[e]], 1.0f);
}

__global__ void invdeg_kernel(const float* __restrict__ deg, float* __restrict__ inv, int n) {
    int i = blockIdx.x * blockDim.x + threadIdx.x;
    if (i < n) inv[i] = 1.0f / fmaxf(deg[i], 1.0f);
}

// one thread per (edge, feature): coalesced gather + atomic scatter (L2-resident)
__global__ void scatter_add_kernel(const float* __restrict__ X, const int* __restrict__ src,
                                   const int* __restrict__ dst, float* __restrict__ AGG, int E) {
    long idx = (long)blockIdx.x * blockDim.x + threadIdx.x;
    if (idx >= (long)E * DIM) return;
    int e = (int)(idx >> 7);
    int f = (int)(idx & (DIM - 1));
    atomicAdd(&AGG[(long)dst[e] * DIM + f], X[(long)src[e] * DIM + f]);
}

// ---------------- SAGE fused GEMM: Y = (AGG*invdeg)@Wl^T + X@Wr^T + bl ----------------
// wave computes a 16-row x 128-col strip with V_WMMA_F32_16X16X4_F32.
// Also accumulates per-column sum / sum-of-squares for train-mode BN.

__launch_bounds__(256)
__global__ void sage_gemm_kernel(const float* __restrict__ X, const float* __restrict__ AGG,
                                 const float* __restrict__ invdeg,
                                 const float* __restrict__ Wl, const float* __restrict__ bl,
                                 const float* __restrict__ Wr,
                                 float* __restrict__ Y, float* __restrict__ colsum,
                                 float* __restrict__ colsq, int nrows) {
    const int lane  = threadIdx.x & 31;
    const int wave  = threadIdx.x >> 5;
    const int row0  = (blockIdx.x * 8 + wave) * 16;
    if (row0 >= nrows) return;                       // wave-uniform: EXEC stays all-1s

    const int mrow  = lane & 15;                     // A: M index / B: N sub-index
    const int khalf = (lane >> 4) << 1;              // 0 for lanes 0-15, 2 for lanes 16-31
    const int arow  = row0 + mrow;
    const float s   = invdeg[arow];

    v8f acc[8] = {};

    const float* __restrict__ aggRow = AGG + (long)arow * DIM;
    const float* __restrict__ xRow   = X   + (long)arow * DIM;

#pragma unroll 1
    for (int k0 = 0; k0 < DIM; k0 += 4) {
        const int kk = k0 + khalf;
        v2f a1 = *(const v2f*)(aggRow + kk);
        a1.x *= s; a1.y *= s;
        v2f a2 = *(const v2f*)(xRow + kk);
#pragma unroll
        for (int t = 0; t < 8; ++t) {
            const int n = t * 16 + mrow;             // B column for this lane
            v2f b1 = *(const v2f*)(Wl + (long)n * DIM + kk);  // B[k][n] = Wl[n][k]
            v2f b2 = *(const v2f*)(Wr + (long)n * DIM + kk);
            acc[t] = __builtin_amdgcn_wmma_f32_16x16x4_f32(false, a1, false, b1,
                                                           (short)0, acc[t], false, false);
            acc[t] = __builtin_amdgcn_wmma_f32_16x16x4_f32(false, a2, false, b2,
                                                           (short)0, acc[t], false, false);
        }
    }

    // epilogue: C/D layout -> lanes<16: rows row0..row0+7, lanes>=16: rows row0+8..row0+15
    const int nlo   = lane & 15;
    const int rbase = row0 + ((lane >> 4) << 3);
#pragma unroll
    for (int t = 0; t < 8; ++t) {
        const int n = t * 16 + nlo;
        const float bias = bl[n];
        float psum = 0.f, psq = 0.f;
#pragma unroll
        for (int j = 0; j < 8; ++j) {
            float v = acc[t][j] + bias;
            Y[(long)(rbase + j) * DIM + n] = v;
            psum += v; psq += v * v;
        }
        // lane L and L+16 hold the same output column -> pairwise reduce, one atomic
        psum += __shfl_xor(psum, 16, 32);
        psq  += __shfl_xor(psq, 16, 32);
        if (lane < 16) {
            atomicAdd(&colsum[n], psum);
            atomicAdd(&colsq[n],  psq);
        }
    }
}

// ---------------- BN finalize: fold mean/var/gamma/beta into scale/shift ----------------

__global__ void bn_stats_kernel(const float* __restrict__ colsum, const float* __restrict__ colsq,
                                const float* __restrict__ g, const float* __restrict__ b,
                                float* __restrict__ kscale, float* __restrict__ kshift, float n) {
    int i = threadIdx.x;
    float mean = colsum[i] / n;
    float var  = colsq[i] / n - mean * mean;   // biased (train-mode) variance
    float istd = rsqrtf(var + BN_EPS);
    float sc   = istd * g[i];
    kscale[i] = sc;
    kshift[i] = b[i] - mean * sc;
}

__global__ void bn_prelu_kernel(const float* __restrict__ Y, const float* __restrict__ kscale,
                                const float* __restrict__ kshift, const float* __restrict__ a,
                                float* __restrict__ Xout, long total) {
    long i = (long)blockIdx.x * blockDim.x + threadIdx.x;
    if (i >= total) return;
    int f = (int)(i & (DIM - 1));
    float v = Y[i] * kscale[f] + kshift[f];
    Xout[i] = v >= 0.f ? v : a[0] * v;
}

// ---------------- JK GEMM: H = relu(cat(x1,x2,x3) @ Wjk^T + bjk), Wjk [128,384] ----------------

__launch_bounds__(256)
__global__ void jk_gemm_kernel(const float* __restrict__ X1, const float* __restrict__ X2,
                               const float* __restrict__ X3,
                               const float* __restrict__ Wjk, const float* __restrict__ bjk,
                               float* __restrict__ H, int nrows) {
    const int lane  = threadIdx.x & 31;
    const int wave  = threadIdx.x >> 5;
    const int row0  = (blockIdx.x * 8 + wave) * 16;
    if (row0 >= nrows) return;

    const int mrow  = lane & 15;
    const int khalf = (lane >> 4) << 1;
    const int arow  = row0 + mrow;

    v8f acc[8] = {};
    const float* srcs[3] = {X1, X2, X3};

#pragma unroll
    for (int seg = 0; seg < 3; ++seg) {
        const float* __restrict__ xRow = srcs[seg] + (long)arow * DIM;
        const int kbase = seg * DIM;
#pragma unroll 1
        for (int k0 = 0; k0 < DIM; k0 += 4) {
            const int kk = k0 + khalf;
            v2f a = *(const v2f*)(xRow + kk);
#pragma unroll
            for (int t = 0; t < 8; ++t) {
                const int n = t * 16 + mrow;
                v2f b = *(const v2f*)(Wjk + (long)n * (3 * DIM) + kbase + kk);
                acc[t] = __builtin_amdgcn_wmma_f32_16x16x4_f32(false, a, false, b,
                                                               (short)0, acc[t], false, false);
            }
        }
    }

    const int nlo   = lane & 15;
    const int rbase = row0 + ((lane >> 4) << 3);
#pragma unroll
    for (int t = 0; t < 8; ++t) {
        const int n = t * 16 + nlo;
        const float bias = bjk[n];
#pragma unroll
        for (int j = 0; j < 8; ++j) {
            float v = acc[t][j] + bias;
            H[(long)(rbase + j) * DIM + n] = fmaxf(v, 0.f);
        }
    }
}

// ---------------- pooling + classifier ----------------

__global__ void pool_kernel(const float* __restrict__ H, const int* __restrict__ batch,
                            float* __restrict__ pooled, int nrows) {
    long i = (long)blockIdx.x * blockDim.x + threadIdx.x;
    if (i >= (long)nrows * DIM) return;
    int n = (int)(i >> 7);
    int f = (int)(i & (DIM - 1));
    atomicAdd(&pooled[(long)batch[n] * DIM + f], H[i]);
}

__global__ void gcount_kernel(const int* __restrict__ batch, float* __restrict__ gcnt, int nrows) {
    int i = blockIdx.x * blockDim.x + threadIdx.x;
    if (i < nrows) atomicAdd(&gcnt[batch[i]], 1.0f);
}

__global__ void final_kernel(const float* __restrict__ pooled, const float* __restrict__ gcnt,
                             const float* __restrict__ Wf, const float* __restrict__ bf,
                             float* __restrict__ out) {
    int i = blockIdx.x * blockDim.x + threadIdx.x;
    if (i >= NGR * NCLS) return;
    int g = i / NCLS, c = i % NCLS;
    float inv = 1.0f / fmaxf(gcnt[g], 1.0f);
    float sum = 0.f;
#pragma unroll 4
    for (int k = 0; k < DIM; ++k) sum += pooled[g * DIM + k] * Wf[c * DIM + k];
    out[i] = sum * inv + bf[c];
}

// ---------------- host orchestration ----------------

extern "C" void kernel_launch(void* const* d_in, const int* in_sizes, int n_in,
                              void* d_out, int out_size, void* d_ws, size_t ws_size,
                              hipStream_t stream) {
    const float* x    = (const float*)d_in[0];
    const int*   ei   = (const int*)d_in[1];
    const int*   batch= (const int*)d_in[2];
    const int*   src  = ei;
    const int*   dst  = ei + NE;

    const float* Wl[3] = {(const float*)d_in[3],  (const float*)d_in[9],  (const float*)d_in[15]};
    const float* blv[3]= {(const float*)d_in[4],  (const float*)d_in[10], (const float*)d_in[16]};
    const float* Wr[3] = {(const float*)d_in[5],  (const float*)d_in[11], (const float*)d_in[17]};
    const float* gg[3] = {(const float*)d_in[6],  (const float*)d_in[12], (const float*)d_in[18]};
    const float* bb[3] = {(const float*)d_in[7],  (const float*)d_in[13], (const float*)d_in[19]};
    const float* aa[3] = {(const float*)d_in[8],  (const float*)d_in[14], (const float*)d_in[20]};
    const float* Wjk = (const float*)d_in[21];
    const float* bjk = (const float*)d_in[22];
    const float* Wf  = (const float*)d_in[23];
    const float* bf  = (const float*)d_in[24];
    float* out = (float*)d_out;

    const long FEAT = (long)NN * DIM;       // 6.4M floats
    float* ws     = (float*)d_ws;
    float* agg    = ws;                     // FEAT (also reused as H)
    float* y      = agg   + FEAT;           // FEAT
    float* x1     = y     + FEAT;           // FEAT
    float* x2     = x1    + FEAT;           // FEAT
    float* x3     = x2    + FEAT;           // FEAT
    float* deg    = x3    + FEAT;           // NN
    float* invdeg = deg   + NN;             // NN
    float* colsum = invdeg+ NN;             // DIM
    float* colsq  = colsum+ DIM;            // DIM
    float* kscale = colsq + DIM;            // DIM
    float* kshift = kscale+ DIM;            // DIM
    float* pooled = kshift+ DIM;            // NGR*DIM
    float* gcnt   = pooled+ NGR * DIM;      // NGR

    const int T = 256;
    const int featBlocks    = (int)((FEAT + T - 1) / T);
    const int scatterBlocks = (int)(((long)NE * DIM + T - 1) / T);
    const int gemmBlocks    = (NN / 16 + 7) / 8;   // 3125 row-tiles, 8 waves/block

    // degree (shared by all layers)
    zero_f32<<<(2 * NN + T - 1) / T, T, 0, stream>>>(deg, 2 * NN);  // deg + invdeg
    degree_kernel<<<(NE + T - 1) / T, T, 0, stream>>>(dst, deg, NE);
    invdeg_kernel<<<(NN + T - 1) / T, T, 0, stream>>>(deg, invdeg, NN);

    const float* xin = x;
    float* xout[3] = {x1, x2, x3};
    for (int l = 0; l < 3; ++l) {
        zero_f32<<<featBlocks, T, 0, stream>>>(agg, FEAT);
        zero_f32<<<1, 2 * DIM, 0, stream>>>(colsum, 2 * DIM);        // colsum + colsq
        scatter_add_kernel<<<scatterBlocks, T, 0, stream>>>(xin, src, dst, agg, NE);
        sage_gemm_kernel<<<gemmBlocks, T, 0, stream>>>(xin, agg, invdeg,
                                                       Wl[l], blv[l], Wr[l],
                                                       y, colsum, colsq, NN);
        bn_stats_kernel<<<1, DIM, 0, stream>>>(colsum, colsq, gg[l], bb[l],
                                               kscale, kshift, (float)NN);
        bn_prelu_kernel<<<featBlocks, T, 0, stream>>>(y, kscale, kshift, aa[l], xout[l], FEAT);
        xin = xout[l];
    }

    // JK concat GEMM + relu (H reuses agg buffer; every element written)
    float* H = agg;
    jk_gemm_kernel<<<gemmBlocks, T, 0, stream>>>(x1, x2, x3, Wjk, bjk, H, NN);

    // global mean pool + classifier
    zero_f32<<<1, NGR * DIM / 32 + NGR, 0, stream>>>(pooled, NGR * DIM + NGR);  // pooled + gcnt
    pool_kernel<<<featBlocks, T, 0, stream>>>(H, batch, pooled, NN);
    gcount_kernel<<<(NN + T - 1) / T, T, 0, stream>>>(batch, gcnt, NN);
    final_kernel<<<(NGR * NCLS + T - 1) / T, T, 0, stream>>>(pooled, gcnt, Wf, bf, out);
}